// ClDiceLoss_8014408974963
// MI455X (gfx1250) — compile-verified
//
#include <hip/hip_runtime.h>
#include <math.h>

// ---------------------------------------------------------------------------
// clDice loss, (B,C,D,H,W) = (2,1,192,192,192) f32.
// Memory-bound stencil: whole working set (113 MB) fits MI455X's 192 MB L2,
// so passes run at L2 speed; we minimize per-lane VALU overhead and LDS-tile
// halo amplification. CDNA5 paths: async global->LDS (ASYNCcnt +
// s_wait_asynccnt), global_prefetch_b8, wave32 shuffle reductions.
// No WMMA: min/max pooling is not a contraction.
// ---------------------------------------------------------------------------

#define DIMW 192
#define SLAB (192 * 192)          // 36864
#define NPB  (192 * 192 * 192)    // 7077888 voxels per batch element
#define NBATCH 2
#define NTOT (NBATCH * NPB)       // 14155776
#define RED_BLOCKS 1024

#if defined(__has_builtin)
#if __has_builtin(__builtin_amdgcn_global_load_async_to_lds_b32)
#define USE_ASYNC_LDS 1
#endif
#if __has_builtin(__builtin_amdgcn_s_wait_asynccnt)
#define HAVE_WAIT_ASYNC_BUILTIN 1
#endif
#endif

#ifdef USE_ASYNC_LDS
// clang (HIP mode) prints addrspace(1) as '__device__' and addrspace(3) as
// '__shared__'; the builtin takes (global int*, shared int*, imm, imm).
typedef __attribute__((address_space(1))) int as1_int_t;
typedef __attribute__((address_space(3))) int as3_int_t;

__device__ __forceinline__ void async_wait_zero() {
#ifdef HAVE_WAIT_ASYNC_BUILTIN
  __builtin_amdgcn_s_wait_asynccnt(0);
#else
  asm volatile("s_wait_asynccnt 0" ::: "memory");
#endif
}
#endif

// ---------------------------------------------------------------------------
// Pass 1: m = min( minpool_z3(x), minpool_y3(x), minpool_x3(x) )
// 3D grid: no per-lane div/mod. Block = 64x4 voxels of one z-plane.
// grid(3, 48, 384): z-plane & batch decoded from blockIdx.z (scalar).
// Boundary clamp == -inf-padded max_pool(-x) semantics of the reference.
// ---------------------------------------------------------------------------
__global__ __launch_bounds__(256) void kmin_axes(const float* __restrict__ X,
                                                 float* __restrict__ M) {
  const int tid = threadIdx.x;
  const int x = blockIdx.x * 64 + (tid & 63);
  const int y = blockIdx.y * 4 + (tid >> 6);
  const int b = blockIdx.z / DIMW;          // scalar
  const int z = blockIdx.z - b * DIMW;      // scalar
  const int n = b * NPB + (z * DIMW + y) * DIMW + x;

  // speculative prefetch two z-slabs ahead (gfx1250: global_prefetch_b8)
  if (z < DIMW - 2) __builtin_prefetch(X + n + 2 * SLAB, 0, 0);

  float c  = X[n];
  float vz = c, vy = c, vx = c;
  if (z > 0)        vz = fminf(vz, X[n - SLAB]);
  if (z < DIMW - 1) vz = fminf(vz, X[n + SLAB]);
  if (y > 0)        vy = fminf(vy, X[n - DIMW]);
  if (y < DIMW - 1) vy = fminf(vy, X[n + DIMW]);
  if (x > 0)        vx = fminf(vx, X[n - 1]);
  if (x < DIMW - 1) vx = fminf(vx, X[n + 1]);
  M[n] = fminf(vx, fminf(vy, vz));
}

// ---------------------------------------------------------------------------
// Pass 2: x = relu(x - relu(maxpool3x3x3(m) - m)), in place on x.
// Tile 8x4x64 outputs + halo 1 => LDS tile 10x6x66 f32 (15840 B),
// halo read amplification 3960/2048 = 1.93x. Tile fetched with async
// global->LDS DMA; OOB halo prefilled with -inf (maxpool pad identity).
// ---------------------------------------------------------------------------
#define TLZ 10
#define TLY 6
#define TLX 66
#define TSZ (TLZ * TLY * TLX)   // 3960

__global__ __launch_bounds__(256) void kcontour(const float* __restrict__ M,
                                                float* __restrict__ X) {
  __shared__ float tile[TSZ];

  const int tid = threadIdx.x;
  const int x0  = blockIdx.x * 64;
  const int y0  = blockIdx.y * 4;
  const int bz  = blockIdx.z;          // [0, 48)
  const int b   = bz / 24;
  const int z0  = (bz - b * 24) * 8;
  const int base_b = b * NPB;

  // Phase 1: -inf prefill (maxpool pad identity)
  for (int i = tid; i < TSZ; i += 256) tile[i] = -__builtin_inff();
  __syncthreads();

  // Phase 2: async-DMA in-bounds tile cells into LDS
  for (int i = tid; i < TSZ; i += 256) {
    int lx = i % TLX;
    int tt = i / TLX;
    int ly = tt % TLY;
    int lz = tt / TLY;
    int gx = x0 + lx - 1, gy = y0 + ly - 1, gz = z0 + lz - 1;
    if ((unsigned)gx < (unsigned)DIMW && (unsigned)gy < (unsigned)DIMW &&
        (unsigned)gz < (unsigned)DIMW) {
      int idx = base_b + (gz * DIMW + gy) * DIMW + gx;
#ifdef USE_ASYNC_LDS
      __builtin_amdgcn_global_load_async_to_lds_b32(
          (as1_int_t*)(M + idx), (as3_int_t*)(&tile[i]), 0, 0);
#else
      tile[i] = M[idx];
#endif
    }
  }
#ifdef USE_ASYNC_LDS
  async_wait_zero();
#endif
  __syncthreads();

  // Phase 3: compute. Thread -> (ly = tid>>6, lx = tid&63), 8 z outputs.
  const int lx = tid & 63;
  const int ly = tid >> 6;

  // 3x3 plane max around (ly+1, lx+1) for each of the 10 LDS z-layers
  float pm[TLZ];
#pragma unroll
  for (int z = 0; z < TLZ; ++z) {
    const float* p = &tile[(z * TLY + ly) * TLX + lx];
    float r0 = fmaxf(fmaxf(p[0],       p[1]),       p[2]);
    float r1 = fmaxf(fmaxf(p[TLX],     p[TLX + 1]), p[TLX + 2]);
    float r2 = fmaxf(fmaxf(p[2 * TLX], p[2 * TLX + 1]), p[2 * TLX + 2]);
    pm[z] = fmaxf(r0, fmaxf(r1, r2));
  }

  const int nbase = base_b + (z0 * DIMW + (y0 + ly)) * DIMW + (x0 + lx);
#pragma unroll
  for (int lz = 0; lz < 8; ++lz) {
    float mp = fmaxf(pm[lz], fmaxf(pm[lz + 1], pm[lz + 2]));
    float mc = tile[((lz + 1) * TLY + (ly + 1)) * TLX + (lx + 1)];
    float contour = fmaxf(mp - mc, 0.0f);
    int n = nbase + lz * SLAB;
    X[n] = fmaxf(X[n] - contour, 0.0f);   // center-only read: in-place safe
  }
}

// ---------------------------------------------------------------------------
// Deterministic two-stage reduction: sum(skel*other) and sum(skel)
// ---------------------------------------------------------------------------
__global__ __launch_bounds__(256) void kreduce1(const float* __restrict__ S,
                                                const float* __restrict__ V,
                                                float* __restrict__ partials) {
  const float4* S4 = (const float4*)S;
  const float4* V4 = (const float4*)V;
  const int stride = gridDim.x * 256;
  float sp = 0.0f, ss = 0.0f;
  for (int i = blockIdx.x * 256 + threadIdx.x; i < NTOT / 4; i += stride) {
    float4 a = S4[i];
    float4 v = V4[i];
    sp += a.x * v.x + a.y * v.y + a.z * v.z + a.w * v.w;
    ss += a.x + a.y + a.z + a.w;
  }
  // wave32 tree
  for (int off = 16; off > 0; off >>= 1) {
    sp += __shfl_xor(sp, off, 32);
    ss += __shfl_xor(ss, off, 32);
  }
  __shared__ float wsum[16];
  int wid = threadIdx.x >> 5, lane = threadIdx.x & 31;
  if (lane == 0) { wsum[wid] = sp; wsum[8 + wid] = ss; }
  __syncthreads();
  if (threadIdx.x == 0) {
    float a = 0.0f, bacc = 0.0f;
    for (int i = 0; i < 8; ++i) { a += wsum[i]; bacc += wsum[8 + i]; }
    partials[2 * blockIdx.x]     = a;
    partials[2 * blockIdx.x + 1] = bacc;
  }
}

__global__ __launch_bounds__(256) void kreduce2(const float* __restrict__ partials,
                                                float* __restrict__ sums2) {
  float sp = 0.0f, ss = 0.0f;
  for (int k = threadIdx.x; k < RED_BLOCKS; k += 256) {
    sp += partials[2 * k];
    ss += partials[2 * k + 1];
  }
  for (int off = 16; off > 0; off >>= 1) {
    sp += __shfl_xor(sp, off, 32);
    ss += __shfl_xor(ss, off, 32);
  }
  __shared__ float wsum[16];
  int wid = threadIdx.x >> 5, lane = threadIdx.x & 31;
  if (lane == 0) { wsum[wid] = sp; wsum[8 + wid] = ss; }
  __syncthreads();
  if (threadIdx.x == 0) {
    float a = 0.0f, bacc = 0.0f;
    for (int i = 0; i < 8; ++i) { a += wsum[i]; bacc += wsum[8 + i]; }
    sums2[0] = a;
    sums2[1] = bacc;
  }
}

__global__ void kfinal(const float* __restrict__ sums, float* __restrict__ out) {
  float clrecall = (sums[0] + 1e-12f) / (sums[1] + 1e-12f); // skelT vs pred
  float clacc    = (sums[2] + 1e-12f) / (sums[3] + 1e-12f); // skelP vs target
  float cldice   = 2.0f * clrecall * clacc / (clrecall + clacc);
  out[0] = 1.0f - cldice;
}

// ---------------------------------------------------------------------------
extern "C" void kernel_launch(void* const* d_in, const int* in_sizes, int n_in,
                              void* d_out, int out_size, void* d_ws, size_t ws_size,
                              hipStream_t stream) {
  const float* pred   = (const float*)d_in[0];
  const float* target = (const float*)d_in[1];
  float* out = (float*)d_out;

  float* xbuf     = (float*)d_ws;          // NTOT floats
  float* mbuf     = xbuf + NTOT;           // NTOT floats
  float* sums     = mbuf + NTOT;           // 4 floats
  float* partials = sums + 4;              // 2*RED_BLOCKS floats

  const dim3 mgrid(3, 48, 384);            // 64x4x1 blocks over 192^3 x B=2
  const dim3 cgrid(3, 48, 48);             // 64x4x8 tiles over 192^3 x B=2

  // --- target skeleton -> sums[0..1] = (sum skelT*pred, sum skelT)
  (void)hipMemcpyAsync(xbuf, target, (size_t)NTOT * sizeof(float),
                       hipMemcpyDeviceToDevice, stream);
  for (int it = 0; it < 5; ++it) {
    kmin_axes<<<mgrid, 256, 0, stream>>>(xbuf, mbuf);
    kcontour<<<cgrid, 256, 0, stream>>>(mbuf, xbuf);
  }
  kreduce1<<<RED_BLOCKS, 256, 0, stream>>>(xbuf, pred, partials);
  kreduce2<<<1, 256, 0, stream>>>(partials, sums);

  // --- pred skeleton -> sums[2..3] = (sum skelP*target, sum skelP)
  (void)hipMemcpyAsync(xbuf, pred, (size_t)NTOT * sizeof(float),
                       hipMemcpyDeviceToDevice, stream);
  for (int it = 0; it < 5; ++it) {
    kmin_axes<<<mgrid, 256, 0, stream>>>(xbuf, mbuf);
    kcontour<<<cgrid, 256, 0, stream>>>(mbuf, xbuf);
  }
  kreduce1<<<RED_BLOCKS, 256, 0, stream>>>(xbuf, target, partials);
  kreduce2<<<1, 256, 0, stream>>>(partials, sums + 2);

  kfinal<<<1, 1, 0, stream>>>(sums, out);
}